// SS3D_v5_34926674051367
// MI455X (gfx1250) — compile-verified
//
#include <hip/hip_runtime.h>
#include <math.h>

#define L4096 4096
#define DI    128      // d_inner
#define NST   16       // d_state
#define KDIR  8        // scan directions
#define NBATCH 2
#define RANK  8        // dt_rank
#define CROWS 40       // dt_rank + 2*d_state
#define CHUNK 64       // scan steps staged per LDS refill

typedef __attribute__((ext_vector_type(2))) float v2f;
typedef __attribute__((ext_vector_type(8))) float v8f;

// ---------------------------------------------------------------------------
// Scan-order gather maps for the 4 rotated views (np.rot90 algebra, S=16):
//   k0: (h,w,p)   k1: (p,15-h,15-w)   k2: (15-h,w,15-p)   k3: (15-p,15-h,w)
// ---------------------------------------------------------------------------
__global__ void build_maps_kernel(const int* __restrict__ order,
                                  int* __restrict__ enc, int* __restrict__ inv) {
    int l = blockIdx.x * blockDim.x + threadIdx.x;
    if (l >= L4096) return;
    int q = order[l];
    int h = (q >> 8) & 15, w = (q >> 4) & 15, p = q & 15;
    int s0 = q;
    int s1 = (p << 8) | ((15 - h) << 4) | (15 - w);
    int s2 = ((15 - h) << 8) | (w << 4) | (15 - p);
    int s3 = ((15 - p) << 8) | ((15 - h) << 4) | w;
    enc[0 * L4096 + l] = s0; inv[0 * L4096 + s0] = l;
    enc[1 * L4096 + l] = s1; inv[1 * L4096 + s1] = l;
    enc[2 * L4096 + l] = s2; inv[2 * L4096 + s2] = l;
    enc[3 * L4096 + l] = s3; inv[3 * L4096 + s3] = l;
}

// ---------------------------------------------------------------------------
// fp32 WMMA GEMM, fully specialized: C(MxN) = A(MxK)*B(KxN), one wave per
// 16x16 tile, K stepped by 4 with V_WMMA_F32_16X16X4_F32, K-loop fully
// unrolled (KC/4 wmma ops per wave, immediate-offset loads).
// A row clamped branchlessly when MGUARD (M=40 case); OOB rows not stored.
// TRANSB: B element (k,n) read from B[n*LDB + k] (weights stored (N,K)).
// ---------------------------------------------------------------------------
template<int KC, int LDA, int LDB, int LDC, bool TRANSB, bool MGUARD>
__global__ __launch_bounds__(32)
void wmma_gemm(const float* __restrict__ A, const float* __restrict__ B,
               float* __restrict__ C, int M, long sA, long sB, long sC) {
    A += blockIdx.z * sA;  B += blockIdx.z * sB;  C += blockIdx.z * sC;

    const int lane  = threadIdx.x;      // 0..31
    const int sub   = lane & 15;        // M index (A) / N index (B, C)
    const int khalf = lane >> 4;        // K pair select {0,1} vs {2,3}
    const int m0    = blockIdx.y * 16;
    const int n0    = blockIdx.x * 16;

    int ra = m0 + sub;
    if (MGUARD) ra = (ra < M) ? ra : (M - 1);          // branchless clamp
    const v2f* __restrict__ A2  = (const v2f*)(A + (long)ra * LDA);
    const int  cb               = n0 + sub;
    const v2f* __restrict__ B2t = (const v2f*)(B + (long)cb * LDB);
    const float* __restrict__ Bn = B + cb;

    v8f acc = {};
#pragma unroll
    for (int i = 0; i < KC / 4; ++i) {
        v2f a = A2[2 * i + khalf];                     // A: K = 4i+2*khalf (+1)
        v2f b;
        if (TRANSB) {
            b = B2t[2 * i + khalf];
        } else {
            const float* bp = Bn + (long)(4 * i + 2 * khalf) * LDB;
            b.x = bp[0];
            b.y = bp[LDB];
        }
        acc = __builtin_amdgcn_wmma_f32_16x16x4_f32(
            false, a, false, b, (short)0, acc, false, false);
    }

    // C/D layout: VGPR r -> row m0+r (lanes 0-15) / m0+8+r (lanes 16-31)
    const int mb = m0 + khalf * 8;
#pragma unroll
    for (int r = 0; r < 8; ++r) {
        const int mr = mb + r;
        if (!MGUARD || mr < M) C[(long)mr * LDC + cb] = acc[r];
    }
}

// ---------------------------------------------------------------------------
// Depthwise 3x3x3 conv (SAME) + bias + SiLU.  xin:(B,s,d) -> xc:(B,d,s)
// ---------------------------------------------------------------------------
__global__ void conv_silu_kernel(const float* __restrict__ xin,
                                 const float* __restrict__ cw,
                                 const float* __restrict__ cb,
                                 float* __restrict__ xc) {
    int t = blockIdx.x * blockDim.x + threadIdx.x;
    if (t >= NBATCH * DI * L4096) return;
    const int d = t & (DI - 1);
    const int s = (t >> 7) & (L4096 - 1);
    const int b = t >> 19;
    const int h = s >> 8, w = (s >> 4) & 15, p = s & 15;
    float acc = cb[d];
#pragma unroll
    for (int kz = 0; kz < 3; ++kz) {
        const int hh = h + kz - 1;
        if (hh < 0 || hh > 15) continue;
#pragma unroll
        for (int ky = 0; ky < 3; ++ky) {
            const int ww = w + ky - 1;
            if (ww < 0 || ww > 15) continue;
#pragma unroll
            for (int kx = 0; kx < 3; ++kx) {
                const int pp = p + kx - 1;
                if (pp < 0 || pp > 15) continue;
                const int sp = (hh << 8) | (ww << 4) | pp;
                acc += cw[((d * 3 + kz) * 3 + ky) * 3 + kx] *
                       xin[((long)b * L4096 + sp) * DI + d];
            }
        }
    }
    const float sv = acc / (1.0f + __expf(-acc));   // SiLU
    xc[((long)(b * DI + d)) * L4096 + s] = sv;
}

// ---------------------------------------------------------------------------
// Build xs (B,8,d,L): k<4 gather via enc map; k>=4 reversed traversal.
// ---------------------------------------------------------------------------
__global__ void build_xs_kernel(const float* __restrict__ xc,
                                const int* __restrict__ enc,
                                float* __restrict__ xs) {
    int t = blockIdx.x * blockDim.x + threadIdx.x;
    if (t >= NBATCH * KDIR * DI * L4096) return;
    const int l = t & (L4096 - 1);
    const int d = (t >> 12) & (DI - 1);
    const int k = (t >> 19) & (KDIR - 1);
    const int b = t >> 22;
    const int kk = k & 3;
    const int ll = (k < 4) ? l : (L4096 - 1 - l);
    const int s  = enc[kk * L4096 + ll];
    xs[t] = xc[((long)(b * DI + d)) * L4096 + s];
}

// ---------------------------------------------------------------------------
// Selective scan: one block per (b,k); thread = channel d; h[16] in VGPRs.
// B/C (32 contiguous rows of x_dbl: rows 8..39) staged per 64-step chunk
// into LDS via CDNA5 async global->LDS DMA (global_load_async_to_lds_b32,
// ASYNCcnt), u/dt streamed as b128, output stored as b128.
// ---------------------------------------------------------------------------
__global__ __launch_bounds__(DI)
void scan_kernel(const float* __restrict__ xs,
                 const float* __restrict__ dts,
                 const float* __restrict__ xdbl,
                 const float* __restrict__ A_logs,
                 const float* __restrict__ Ds,
                 const float* __restrict__ dtb,
                 float* __restrict__ out_y) {
    const int bk = blockIdx.x;           // b*8 + k
    const int k  = bk & (KDIR - 1);
    const int d  = threadIdx.x;

    const float* u_row   = xs  + ((long)bk * DI + d) * L4096;
    const float* dt_row  = dts + ((long)bk * DI + d) * L4096;
    const float* BCrows  = xdbl + ((long)bk * CROWS + RANK) * (long)L4096; // 32 rows: B(16),C(16)
    float*       out_row = out_y + ((long)bk * DI + d) * L4096;

    float A[NST];
#pragma unroll
    for (int n = 0; n < NST; ++n)
        A[n] = -__expf(A_logs[((long)(k * DI + d)) * NST + n]);
    const float Dv   = Ds[k * DI + d];
    const float bias = dtb[k * DI + d];

    float h[NST];
#pragma unroll
    for (int n = 0; n < NST; ++n) h[n] = 0.0f;

    __shared__ float sBC[32 * CHUNK];    // [n(32)][lc(64)]; B = n<16, C = n>=16
    float* const sB = sBC;
    float* const sC = sBC + NST * CHUNK;

    for (int l0 = 0; l0 < L4096; l0 += CHUNK) {
        // ---- async-stage the 32xCHUNK B/C tile into LDS ----
#pragma unroll
        for (int j = 0; j < (32 * CHUNK) / DI; ++j) {     // 16 per thread
            const int idx = threadIdx.x + j * DI;
            const int n   = idx >> 6;                     // row 0..31
            const int lc  = idx & (CHUNK - 1);
            const float* g = BCrows + (long)n * L4096 + l0 + lc;
            const unsigned lds_addr = (unsigned)(unsigned long long)(&sBC[idx]);
            const unsigned long long ga = (unsigned long long)g;
            asm volatile("global_load_async_to_lds_b32 %0, %1, off"
                         :: "v"(lds_addr), "v"(ga) : "memory");
        }
        asm volatile("s_wait_asynccnt 0x0" ::: "memory");
        __syncthreads();

        // prefetch next chunk of this thread's u/dt stream (speculative ok)
        __builtin_prefetch(u_row + l0 + CHUNK, 0, 1);
        __builtin_prefetch(dt_row + l0 + CHUNK, 0, 1);

        // ---- consume CHUNK steps ----
        for (int lq = 0; lq < CHUNK; lq += 4) {
            const float4 uq = *(const float4*)(u_row + l0 + lq);
            const float4 dq = *(const float4*)(dt_row + l0 + lq);
            float yv[4];
            const float uu[4] = {uq.x, uq.y, uq.z, uq.w};
            const float dd[4] = {dq.x, dq.y, dq.z, dq.w};
#pragma unroll
            for (int e = 0; e < 4; ++e) {
                const int lc = lq + e;
                const float dtraw = dd[e] + bias;
                const float dt_s  = (dtraw > 20.0f) ? dtraw
                                  : __logf(1.0f + __expf(dtraw));   // softplus
                const float du = dt_s * uu[e];
                float y = 0.0f;
#pragma unroll
                for (int n = 0; n < NST; ++n) {
                    const float dA = __expf(dt_s * A[n]);
                    h[n] = dA * h[n] + du * sB[n * CHUNK + lc];
                    y += h[n] * sC[n * CHUNK + lc];
                }
                yv[e] = y + Dv * uu[e];
            }
            float4 yo; yo.x = yv[0]; yo.y = yv[1]; yo.z = yv[2]; yo.w = yv[3];
            *(float4*)(out_row + l0 + lq) = yo;
        }
        __syncthreads();
    }
}

// ---------------------------------------------------------------------------
// Decode + accumulate 8 directions -> y (b,s,d).
// ---------------------------------------------------------------------------
__global__ void combine_kernel(const float* __restrict__ out_y,
                               const int* __restrict__ inv,
                               float* __restrict__ y) {
    int t = blockIdx.x * blockDim.x + threadIdx.x;
    if (t >= NBATCH * L4096 * DI) return;
    const int d = t & (DI - 1);
    const int s = (t >> 7) & (L4096 - 1);
    const int b = t >> 19;
    float acc = 0.0f;
#pragma unroll
    for (int k = 0; k < 4; ++k) {
        const int l = inv[k * L4096 + s];
        acc += out_y[((long)((b * KDIR + k) * DI) + d) * L4096 + l];
        acc += out_y[((long)((b * KDIR + k + 4) * DI) + d) * L4096 + (L4096 - 1 - l)];
    }
    y[t] = acc;
}

// ---------------------------------------------------------------------------
// LayerNorm over channels (128). One block per (b,s).
// ---------------------------------------------------------------------------
__global__ __launch_bounds__(DI)
void layernorm_kernel(float* __restrict__ y,
                      const float* __restrict__ gamma,
                      const float* __restrict__ beta) {
    __shared__ float red[DI];
    const int bs = blockIdx.x;
    const int d  = threadIdx.x;
    const float v = y[(long)bs * DI + d];
    red[d] = v; __syncthreads();
    for (int off = DI / 2; off > 0; off >>= 1) {
        if (d < off) red[d] += red[d + off];
        __syncthreads();
    }
    const float mean = red[0] * (1.0f / DI);
    __syncthreads();
    const float c = v - mean;
    red[d] = c * c; __syncthreads();
    for (int off = DI / 2; off > 0; off >>= 1) {
        if (d < off) red[d] += red[d + off];
        __syncthreads();
    }
    const float var = red[0] * (1.0f / DI);
    y[(long)bs * DI + d] = c * rsqrtf(var + 1e-5f) * gamma[d] + beta[d];
}

// ---------------------------------------------------------------------------
extern "C" void kernel_launch(void* const* d_in, const int* in_sizes, int n_in,
                              void* d_out, int out_size, void* d_ws, size_t ws_size,
                              hipStream_t stream) {
    (void)in_sizes; (void)n_in; (void)out_size; (void)ws_size;

    const float* x       = (const float*)d_in[0];   // (2,16,16,16,128)
    const float* in_w    = (const float*)d_in[1];   // (128,128)
    const float* conv_w  = (const float*)d_in[2];   // (128,1,3,3,3)
    const float* conv_b  = (const float*)d_in[3];   // (128)
    const float* xproj_w = (const float*)d_in[4];   // (8,40,128)
    const float* dtw     = (const float*)d_in[5];   // (8,128,8)
    const float* dtb     = (const float*)d_in[6];   // (8,128)
    const float* A_logs  = (const float*)d_in[7];   // (1024,16)
    const float* Ds      = (const float*)d_in[8];   // (1024)
    const float* gamma   = (const float*)d_in[9];   // (128)
    const float* beta    = (const float*)d_in[10];  // (128)
    const float* out_w   = (const float*)d_in[11];  // (128,128)
    const int*   order   = (const int*)d_in[12];    // (4096)

    float* ws   = (float*)d_ws;
    float* xin  = ws;                                           // (2,4096,128)
    float* xc   = xin  + (size_t)NBATCH * L4096 * DI;           // (2,128,4096)
    float* xs   = xc   + (size_t)NBATCH * L4096 * DI;           // (2,8,128,4096)
    float* xdbl = xs   + (size_t)NBATCH * KDIR * DI * L4096;    // (2,8,40,4096)
    float* dts  = xdbl + (size_t)NBATCH * KDIR * CROWS * L4096; // (2,8,128,4096)
    float* outy = dts  + (size_t)NBATCH * KDIR * DI * L4096;    // (2,8,128,4096)
    float* yc   = outy + (size_t)NBATCH * KDIR * DI * L4096;    // (2,4096,128)
    int*   enc  = (int*)(yc + (size_t)NBATCH * L4096 * DI);     // (4,4096)
    int*   inv  = enc + 4 * L4096;                              // (4,4096)

    const int ROWS = NBATCH * L4096;   // 8192

    build_maps_kernel<<<dim3(L4096 / 256), dim3(256), 0, stream>>>(order, enc, inv);

    // in_proj: xin(8192,128) = x(8192,128) * in_w^T
    wmma_gemm<128, 128, 128, 128, true, false>
        <<<dim3(DI / 16, ROWS / 16, 1), dim3(32), 0, stream>>>(
            x, in_w, xin, ROWS, 0, 0, 0);

    conv_silu_kernel<<<dim3((NBATCH * DI * L4096) / 256), dim3(256), 0, stream>>>(
        xin, conv_w, conv_b, xc);

    build_xs_kernel<<<dim3((NBATCH * KDIR * DI * L4096) / 256), dim3(256), 0, stream>>>(
        xc, enc, xs);

    for (int b = 0; b < NBATCH; ++b) {
        // x_dbl(b,k): (40,4096) = x_proj_w[k](40,128) * xs[b,k](128,4096)
        wmma_gemm<128, 128, L4096, L4096, false, true>
            <<<dim3(L4096 / 16, (CROWS + 15) / 16, KDIR), dim3(32), 0, stream>>>(
                xproj_w,
                xs   + (size_t)b * KDIR * DI * L4096,
                xdbl + (size_t)b * KDIR * CROWS * L4096,
                CROWS,
                (long)CROWS * DI, (long)DI * L4096, (long)CROWS * L4096);
        // dts(b,k): (128,4096) = dt_projs_w[k](128,8) * dtr[b,k](8,4096)
        wmma_gemm<RANK, RANK, L4096, L4096, false, false>
            <<<dim3(L4096 / 16, DI / 16, KDIR), dim3(32), 0, stream>>>(
                dtw,
                xdbl + (size_t)b * KDIR * CROWS * L4096,   // rows 0..7 = dtr
                dts  + (size_t)b * KDIR * DI * L4096,
                DI,
                (long)DI * RANK, (long)CROWS * L4096, (long)DI * L4096);
    }

    scan_kernel<<<dim3(NBATCH * KDIR), dim3(DI), 0, stream>>>(
        xs, dts, xdbl, A_logs, Ds, dtb, outy);

    combine_kernel<<<dim3((NBATCH * L4096 * DI) / 256), dim3(256), 0, stream>>>(
        outy, inv, yc);

    layernorm_kernel<<<dim3(NBATCH * L4096), dim3(DI), 0, stream>>>(yc, gamma, beta);

    // out_proj: d_out(8192,128) = yc(8192,128) * out_w^T
    wmma_gemm<128, 128, 128, 128, true, false>
        <<<dim3(DI / 16, ROWS / 16, 1), dim3(32), 0, stream>>>(
            yc, out_w, (float*)d_out, ROWS, 0, 0, 0);
}